// Net_37752762531949
// MI455X (gfx1250) — compile-verified
//
#include <hip/hip_runtime.h>
#include <hip/hip_bf16.h>
#include <stdint.h>

#define FEATURE_DIM 131072
#define HIDDEN_DIM  512
#define N_CLASSES   524288
#define BATCH       256
#define NNZ_IN      128
#define N_ACTIVE    1024

typedef __attribute__((ext_vector_type(2))) float        v2f;
typedef __attribute__((ext_vector_type(8))) float        v8f;
typedef __attribute__((ext_vector_type(4))) unsigned int u32x4;
typedef __attribute__((ext_vector_type(8))) int          i32x8;
typedef __attribute__((ext_vector_type(4))) int          i32x4;

// ---------------------------------------------------------------------------
// Stage 1: val1[b,h] = relu( sum_n in_values[b,n] * W1[h, idx[b,n]] + b1[h] )
// One block per batch sample; 512 threads = one thread per hidden unit.
// idx/value pairs staged in LDS once and broadcast to all 512 threads.
// ---------------------------------------------------------------------------
__global__ __launch_bounds__(512) void slide_stage1(
    const float* __restrict__ in_values,   // [B, NNZ]
    const int*   __restrict__ in_idx,      // [B, NNZ]
    const float* __restrict__ W1,          // [H, F]
    const float* __restrict__ b1,          // [H]
    float*       __restrict__ val1)        // [B, H] (workspace)
{
    __shared__ float s_val[NNZ_IN];
    __shared__ int   s_idx[NNZ_IN];

    const int b = blockIdx.x;
    const int h = threadIdx.x;

    if (h < NNZ_IN) {
        s_val[h] = in_values[b * NNZ_IN + h];
        s_idx[h] = in_idx[b * NNZ_IN + h];
    }
    __syncthreads();

    const float* __restrict__ row = W1 + (size_t)h * FEATURE_DIM;
    float acc = b1[h];
#pragma unroll 8
    for (int n = 0; n < NNZ_IN; ++n) {
        acc = fmaf(s_val[n], row[s_idx[n]], acc);
    }
    val1[b * HIDDEN_DIM + h] = acc > 0.0f ? acc : 0.0f;
}

// ---------------------------------------------------------------------------
// Stage 2: val2[b,j] = dot(W2[label[b,j],:], val1[b,:]) + b2[label[b,j]]
// Grid (BATCH, 16), 128 threads = 4 waves. Each wave computes 16 labels via a
// chain of V_WMMA_F32_16X16X4_F32: A = 16 gathered W2 rows (K-slice of 4),
// B = val1[kc..kc+3] broadcast to all 16 columns. val1 row is brought to LDS
// by the Tensor Data Mover.
// ---------------------------------------------------------------------------
__global__ __launch_bounds__(128) void slide_stage2(
    const float* __restrict__ val1,     // [B, H] (workspace, f32)
    const int*   __restrict__ labels,   // [B, J]
    const float* __restrict__ W2,       // [C, H]
    const float* __restrict__ b2,       // [C]
    float*       __restrict__ out_val,  // [B, J]
    float*       __restrict__ out_idx)  // [B, J] (indices as float)
{
    __shared__ float s_v1[HIDDEN_DIM];  // 2 KB: this batch's val1 row

    const int b    = blockIdx.x;
    const int t    = threadIdx.x;
    const int lane = t & 31;
    const int wave = t >> 5;

    // --- TDM: DMA val1[b, 0:512] (2KB, one 512x1 f32 tile) into LDS --------
    if (wave == 0) {
        const float* src = val1 + (size_t)b * HIDDEN_DIM;
        unsigned long long ga  = (unsigned long long)(uintptr_t)src;
        unsigned int       lds = (unsigned int)(uintptr_t)(&s_v1[0]);

        // D# group 0: count=1, lds_addr, global_addr[56:0], type=2
        u32x4 g0;
        g0[0] = 1u;                                            // count=1, user mode
        g0[1] = lds;                                           // lds_addr
        g0[2] = (unsigned int)(ga & 0xFFFFFFFFu);              // global_addr[31:0]
        g0[3] = ((unsigned int)(ga >> 32) & 0x01FFFFFFu)       // global_addr[56:32]
                | (2u << 30);                                  // type = 2 ("image")

        // D# group 1: data_size=4B, tensor 512x1, tile 512x1, dim0 stride 512
        i32x8 g1;
        g1[0] = 0x20000;                 // data_size=2 (4 bytes); mask/flags = 0
        g1[1] = (HIDDEN_DIM << 16);      // tensor_dim0[15:0] in bits [31:16]
        g1[2] = (1 << 16);               // tensor_dim0 hi = 0, tensor_dim1 = 1
        g1[3] = (HIDDEN_DIM << 16);      // tensor_dim1 hi = 0, tile_dim0 = 512
        g1[4] = 1;                       // tile_dim1 = 1, tile_dim2 = 0
        g1[5] = HIDDEN_DIM;              // tensor_dim0_stride[31:0] = 512
        g1[6] = 0;                       // stride hi / tensor_dim1_stride = 0
        g1[7] = 0;

        i32x4 gz4 = {0, 0, 0, 0};
        i32x8 gz8 = {0, 0, 0, 0, 0, 0, 0, 0};
        __builtin_amdgcn_tensor_load_to_lds(g0, g1, gz4, gz4, gz8, 0);
        __builtin_amdgcn_s_wait_tensorcnt(0);
    }
    __syncthreads();

    // --- 16 labels per wave ------------------------------------------------
    const int j0 = (blockIdx.y * 4 + wave) * 16;           // label group base
    const int m  = lane & 15;                              // A-matrix row
    const int kh = (lane >> 4) * 2;                        // K sub-pair: 0 or 2

    const int lbl = labels[b * N_ACTIVE + j0 + m];
    const float* __restrict__ rowp = W2 + (size_t)lbl * HIDDEN_DIM;

    v8f c = {};
#pragma unroll 4
    for (int kc = 0; kc < HIDDEN_DIM; kc += 4) {
        // A tile (16x4 f32): lane<16 holds A[m, kc..kc+1], lane>=16 A[m, kc+2..kc+3]
        const float2 a  = *(const float2*)(rowp + kc + kh);
        // B tile (4x16 f32): val1 K-pair broadcast across all 16 columns
        const float2 bb = *(const float2*)(&s_v1[kc + kh]);
        v2f A;  A[0]  = a.x;  A[1]  = a.y;
        v2f Bv; Bv[0] = bb.x; Bv[1] = bb.y;
        c = __builtin_amdgcn_wmma_f32_16x16x4_f32(
                /*neg_a=*/false, A, /*neg_b=*/false, Bv,
                /*c_mod=*/(short)0, c, /*reuse_a=*/false, /*reuse_b=*/false);
    }

    // D layout: lane 0 holds rows M=0..7 (col 0) in c[0..7]; lane 16 rows 8..15.
    // All 16 columns are identical (B was a broadcast), so col 0 suffices.
    if (lane == 0 || lane == 16) {
        const int mbase = (lane >> 4) * 8;
#pragma unroll
        for (int r = 0; r < 8; ++r) {
            const int j = j0 + mbase + r;
            const int l = labels[b * N_ACTIVE + j];
            out_val[b * N_ACTIVE + j] = c[r] + b2[l];
        }
    }
    if (lane == 1 || lane == 17) {
        const int mbase = (lane >> 4) * 8;
#pragma unroll
        for (int r = 0; r < 8; ++r) {
            const int j = j0 + mbase + r;
            out_idx[b * N_ACTIVE + j] = (float)labels[b * N_ACTIVE + j];
        }
    }
}

// ---------------------------------------------------------------------------
extern "C" void kernel_launch(void* const* d_in, const int* in_sizes, int n_in,
                              void* d_out, int out_size, void* d_ws, size_t ws_size,
                              hipStream_t stream) {
    const float* in_values = (const float*)d_in[0];   // [B, NNZ]
    const int*   in_idx    = (const int*)  d_in[1];   // [B, NNZ]
    const int*   labels    = (const int*)  d_in[2];   // [B, J]
    const float* W1        = (const float*)d_in[3];   // [H, F]
    const float* b1        = (const float*)d_in[4];   // [H]
    const float* W2        = (const float*)d_in[5];   // [C, H]
    const float* b2        = (const float*)d_in[6];   // [C]

    float* val1    = (float*)d_ws;                    // [B, H] scratch (512 KB)
    float* out_val = (float*)d_out;                   // [B, J]
    float* out_idx = out_val + (size_t)BATCH * N_ACTIVE;

    slide_stage1<<<dim3(BATCH), dim3(HIDDEN_DIM), 0, stream>>>(
        in_values, in_idx, W1, b1, val1);

    slide_stage2<<<dim3(BATCH, N_ACTIVE / 64), dim3(128), 0, stream>>>(
        val1, labels, W2, b2, out_val, out_idx);
}